// MultiHashEncoding_24223615550002
// MI455X (gfx1250) — compile-verified
//
#include <hip/hip_runtime.h>
#include <stdint.h>

#define N_SAMPLES   262144
#define HASH_MAP_SZ (1u << 21)
#define HASH_MASK   (HASH_MAP_SZ - 1u)
#define PRECOND     10.0f
#define BLOCK       256

__global__ __launch_bounds__(BLOCK) void mhenc_kernel(
    const float* __restrict__ xs,
    const float* __restrict__ g0, const float* __restrict__ g1,
    const float* __restrict__ g2, const float* __restrict__ g3,
    const float* __restrict__ g4, const float* __restrict__ g5,
    const float* __restrict__ g6,
    const float* __restrict__ hv,
    float*       __restrict__ out)
{
    __shared__ float s_xs[BLOCK * 3];
    const int tid = threadIdx.x;
    const int i   = blockIdx.x * BLOCK + tid;

    // ---- Stage this block's xs (256x3 f32 = 3 KB) into LDS via gfx1250
    // ---- async-to-LDS loads (ASYNCcnt), then barrier, then strided ds reads.
    {
        const uint32_t lbase = (uint32_t)(uintptr_t)(&s_xs[0]); // LDS offset = low 32 bits of flat addr
        const float*   gbase = xs + (size_t)blockIdx.x * (BLOCK * 3) + tid;
        #pragma unroll
        for (int k = 0; k < 3; ++k) {
            uint32_t loff = lbase + (uint32_t)(tid + k * BLOCK) * 4u;
            uint64_t ga   = (uint64_t)(uintptr_t)(gbase + k * BLOCK);
            asm volatile("global_load_async_to_lds_b32 %0, %1, off"
                         :: "v"(loff), "v"(ga) : "memory");
        }
        asm volatile("s_wait_asynccnt 0" ::: "memory");
        __syncthreads();
    }

    const float x = s_xs[tid * 3 + 0];
    const float y = s_xs[tid * 3 + 1];
    const float z = s_xs[tid * 3 + 2];

    // xs_n = (x - WORLD_MIN) / (WORLD_MAX - WORLD_MIN), WORLD = [-2, 2]
    const float xn = (x + 2.0f) * 0.25f;
    const float yn = (y + 2.0f) * 0.25f;
    const float zn = (z + 2.0f) * 0.25f;

    // NOTE: idxs is provably irrelevant: every indexed table dim has extent 1
    // and XLA gathers clamp OOB indices, so the effective index is always 0.
    // Dropping it keeps every table base uniform (SGPR pair) so gathers lower
    // to saddr + 32-bit voffset + scale_offset form.

    float* __restrict__ op = out + (size_t)i * 40u;

    // ------------------------- dense grid levels -------------------------
    const int GS[7] = {16, 22, 31, 43, 59, 83, 115};
    const float* gptr[7] = {g0, g1, g2, g3, g4, g5, g6};

    #pragma unroll
    for (int s = 0; s < 7; ++s) {
        const int   g  = GS[s];
        const float gf = (float)g;
        const float lx = xn * gf - 0.5f;
        const float ly = yn * gf - 0.5f;
        const float lz = zn * gf - 0.5f;
        const float fx = floorf(lx), fy = floorf(ly), fz = floorf(lz);
        const float tx = lx - fx,   ty = ly - fy,   tz = lz - fz;
        const int   cx = (int)fx,   cy = (int)fy,   cz = (int)fz;
        const int x0 = min(max(cx,     0), g - 1), x1 = min(max(cx + 1, 0), g - 1);
        const int y0 = min(max(cy,     0), g - 1), y1 = min(max(cy + 1, 0), g - 1);
        const int z0 = min(max(cz,     0), g - 1), z1 = min(max(cz + 1, 0), g - 1);
        const float wx0 = 1.0f - tx, wy0 = 1.0f - ty, wz0 = 1.0f - tz;

        // one base index + three deltas instead of per-corner 2-mad chains
        const uint32_t b000 = (uint32_t)((x0 * g + y0) * g + z0);
        const uint32_t dX   = (uint32_t)((x1 - x0) * g * g);
        const uint32_t dY   = (uint32_t)((y1 - y0) * g);
        const uint32_t dZ   = (uint32_t)(z1 - z0);

        const float wxy00 = wx0 * wy0, wxy01 = wx0 * ty;
        const float wxy10 = tx  * wy0, wxy11 = tx  * ty;

        const float2* __restrict__ gb = reinterpret_cast<const float2*>(gptr[s]);
        float a0 = 0.0f, a1 = 0.0f;
        #pragma unroll
        for (int o = 0; o < 8; ++o) {
            const uint32_t e = b000 + ((o & 4) ? dX : 0u)
                                    + ((o & 2) ? dY : 0u)
                                    + ((o & 1) ? dZ : 0u);
            const float wxy = (o & 4) ? ((o & 2) ? wxy11 : wxy10)
                                      : ((o & 2) ? wxy01 : wxy00);
            const float w = wxy * ((o & 1) ? tz : wz0);
            const float2 v = gb[e];
            a0 = fmaf(v.x, w, a0);
            a1 = fmaf(v.y, w, a1);
        }
        float2 r; r.x = a0 * PRECOND; r.y = a1 * PRECOND;
        *reinterpret_cast<float2*>(op + 2 * s) = r;
    }

    // -------------------------- hashed levels ---------------------------
    const int HS[13] = {159, 221, 307, 427, 592, 823, 1142,
                        1586, 2203, 3059, 4248, 5899, 8192};

    #pragma unroll
    for (int l = 0; l < 13; ++l) {
        const float hf = (float)HS[l];
        const float lx = xn * hf, ly = yn * hf, lz = zn * hf;
        const float fx = floorf(lx), fy = floorf(ly), fz = floorf(lz);
        const float tx = lx - fx,   ty = ly - fy,   tz = lz - fz;
        const int   cx = (int)fx,   cy = (int)fy,   cz = (int)fz;
        const float wx0 = 1.0f - tx, wy0 = 1.0f - ty, wz0 = 1.0f - tz;

        // int32 wraparound multiplies + floor-mod by 2^21 == uint mul + mask
        const uint32_t hx0 = (uint32_t)cx;
        const uint32_t hx1 = (uint32_t)(cx + 1);
        const uint32_t hy0 = (uint32_t)cy       * 19349663u;
        const uint32_t hy1 = (uint32_t)(cy + 1) * 19349663u;
        const uint32_t hz0 = (uint32_t)cz       * 83492791u;
        const uint32_t hz1 = (uint32_t)(cz + 1) * 83492791u;

        const float wxy00 = wx0 * wy0, wxy01 = wx0 * ty;
        const float wxy10 = tx  * wy0, wxy11 = tx  * ty;

        const float2* __restrict__ hb =
            reinterpret_cast<const float2*>(hv) + (size_t)l * HASH_MAP_SZ;
        float a0 = 0.0f, a1 = 0.0f;
        #pragma unroll
        for (int o = 0; o < 8; ++o) {
            const uint32_t e = (((o & 4) ? hx1 : hx0) ^
                                ((o & 2) ? hy1 : hy0) ^
                                ((o & 1) ? hz1 : hz0)) & HASH_MASK;
            const float wxy = (o & 4) ? ((o & 2) ? wxy11 : wxy10)
                                      : ((o & 2) ? wxy01 : wxy00);
            const float w = wxy * ((o & 1) ? tz : wz0);
            const float2 v = hb[e];
            a0 = fmaf(v.x, w, a0);
            a1 = fmaf(v.y, w, a1);
        }
        float2 r; r.x = a0 * PRECOND; r.y = a1 * PRECOND;
        *reinterpret_cast<float2*>(op + 14 + 2 * l) = r;
    }
}

extern "C" void kernel_launch(void* const* d_in, const int* in_sizes, int n_in,
                              void* d_out, int out_size, void* d_ws, size_t ws_size,
                              hipStream_t stream) {
    // d_in[0] = idxs (unused: extent-1 dims + XLA clamp semantics => always 0)
    const float* xs   = (const float*)d_in[1];
    const float* g0   = (const float*)d_in[2];
    const float* g1   = (const float*)d_in[3];
    const float* g2   = (const float*)d_in[4];
    const float* g3   = (const float*)d_in[5];
    const float* g4   = (const float*)d_in[6];
    const float* g5   = (const float*)d_in[7];
    const float* g6   = (const float*)d_in[8];
    const float* hv   = (const float*)d_in[9];
    float*       out  = (float*)d_out;

    dim3 grid(N_SAMPLES / BLOCK), block(BLOCK);
    hipLaunchKernelGGL(mhenc_kernel, grid, block, 0, stream,
                       xs, g0, g1, g2, g3, g4, g5, g6, hv, out);
}